// Attention_axial_24464133718789
// MI455X (gfx1250) — compile-verified
//
#include <hip/hip_runtime.h>
#include <math.h>

typedef __attribute__((ext_vector_type(2))) float v2f;
typedef __attribute__((ext_vector_type(8))) float v8f;

#define HSP   64
#define WSP   4
#define NTOK  256
#define HEADS 6
#define HD    32
#define DIM   192
#define BATCH 16
#define POSD  12
#define PTAB_ROWS 889          // (2*HSP-1)*(2*WSP-1) = 127*7
#define SCALE 0.17677669529663687f  // 32^-0.5

// window token n -> flat L index inside (H=64, W=64) image, window column w (0..15)
__device__ __forceinline__ int lidx(int n, int w) {
  return ((n >> 2) << 6) + (w << 2) + (n & 3);
}

// ---------------------------------------------------------------------------
// Kernel 1: dynamic relative position bias MLP  ->  ptab[889][6] in workspace
// ---------------------------------------------------------------------------
__device__ __forceinline__ void ln_relu12(const float* x, const float* g,
                                          const float* b, float* y) {
  float m = 0.0f;
#pragma unroll
  for (int j = 0; j < POSD; ++j) m += x[j];
  m *= (1.0f / POSD);
  float v = 0.0f;
#pragma unroll
  for (int j = 0; j < POSD; ++j) { float d = x[j] - m; v += d * d; }
  v *= (1.0f / POSD);
  float inv = rsqrtf(v + 1e-5f);
#pragma unroll
  for (int j = 0; j < POSD; ++j) {
    float z = (x[j] - m) * inv * g[j] + b[j];
    y[j] = z > 0.0f ? z : 0.0f;
  }
}

__global__ void bias_mlp_kernel(const float* __restrict__ pw, const float* __restrict__ pb,
                                const float* __restrict__ g1, const float* __restrict__ b1,
                                const float* __restrict__ w1, const float* __restrict__ fb1,
                                const float* __restrict__ g2, const float* __restrict__ b2,
                                const float* __restrict__ w2, const float* __restrict__ fb2,
                                const float* __restrict__ g3, const float* __restrict__ b3,
                                const float* __restrict__ w3, const float* __restrict__ fb3,
                                float* __restrict__ ptab) {
  int i = blockIdx.x * blockDim.x + threadIdx.x;
  if (i >= PTAB_ROWS) return;
  float gh = (float)(i / 7) - 63.0f;   // arange(1-HSP, HSP)
  float gw = (float)(i % 7) - 3.0f;    // arange(1-WSP, WSP)

  float t[POSD], u[POSD];
#pragma unroll
  for (int j = 0; j < POSD; ++j) t[j] = gh * pw[j] + gw * pw[POSD + j] + pb[j];

  ln_relu12(t, g1, b1, u);
#pragma unroll
  for (int j = 0; j < POSD; ++j) {
    float a = fb1[j];
#pragma unroll
    for (int k = 0; k < POSD; ++k) a += u[k] * w1[k * POSD + j];
    t[j] = a;
  }
  ln_relu12(t, g2, b2, u);
#pragma unroll
  for (int j = 0; j < POSD; ++j) {
    float a = fb2[j];
#pragma unroll
    for (int k = 0; k < POSD; ++k) a += u[k] * w2[k * POSD + j];
    t[j] = a;
  }
  ln_relu12(t, g3, b3, u);
#pragma unroll
  for (int h = 0; h < HEADS; ++h) {
    float a = fb3[h];
#pragma unroll
    for (int k = 0; k < POSD; ++k) a += u[k] * w3[k * HEADS + h];
    ptab[i * HEADS + h] = a;
  }
}

// ---------------------------------------------------------------------------
// Kernel 2: windowed attention, one (window, head) per block.
// block = 128 threads = 4 waves; each wave handles 4 query tiles of 16 rows.
// FP32 WMMA (v_wmma_f32_16x16x4_f32) for both QK^T and PV.
// K/V staged to LDS with GLOBAL_LOAD_ASYNC_TO_LDS_B128 (ASYNCcnt-tracked).
// ---------------------------------------------------------------------------
#define KV_STRIDE 36     // 32 + pad, keeps 16B alignment & conflict-free cols
#define S_STRIDE  258    // 256 + pad, even -> aligned b64 A-operand reads
#define KV_FLOATS (NTOK * KV_STRIDE)          // 9216
#define S_FLOATS  (16 * S_STRIDE)             // 4128 per wave

__global__ void attn_kernel(const float* __restrict__ qkv,
                            const float* __restrict__ ptab,
                            float* __restrict__ out) {
  extern __shared__ float smem[];
  float* Ks = smem;                       // [256][36]
  float* Vs = smem + KV_FLOATS;           // [256][36]
  const int lane = threadIdx.x & 31;
  const int wave = threadIdx.x >> 5;
  const int half = lane >> 4;             // 0: lanes 0-15, 1: lanes 16-31
  const int l16  = lane & 15;
  float* Ss = smem + 2 * KV_FLOATS + wave * S_FLOATS;   // per-wave [16][258]

  const int win = blockIdx.x;             // 0..255
  const int hh  = blockIdx.y;             // 0..5
  const int bb  = win >> 4;
  const int w   = win & 15;

  const size_t selStride = (size_t)BATCH * 4096 * DIM;  // q/k/v plane stride
  const float* qb = qkv + (size_t)bb * 4096 * DIM + hh * HD;
  const float* kb = qb + selStride;
  const float* vb = qb + 2 * selStride;

  // ---- stage K and V into LDS asynchronously (memory -> LDS, no VGPR bounce)
  for (int i = threadIdx.x; i < 2048; i += 128) {      // 256 rows * 8 float4
    int row = i >> 3;
    int f4  = (i & 7) << 2;
    int g   = lidx(row, w) * DIM + f4;
    unsigned int koff = (unsigned int)((row * KV_STRIDE + f4) * 4);
    unsigned int voff = koff + (unsigned int)(KV_FLOATS * 4);
    unsigned long long ka = (unsigned long long)(uintptr_t)(kb + g);
    unsigned long long va = (unsigned long long)(uintptr_t)(vb + g);
    asm volatile("global_load_async_to_lds_b128 %0, %1, off"
                 :: "v"(koff), "v"(ka) : "memory");
    asm volatile("global_load_async_to_lds_b128 %0, %1, off"
                 :: "v"(voff), "v"(va) : "memory");
  }
  asm volatile("s_wait_asynccnt 0x0" ::: "memory");
  __syncthreads();

  for (int qt = wave; qt < 16; qt += 4) {
    const int qrow0 = qt << 4;

    // ---- Q tile in WMMA A-layout registers, pre-scaled (streaming reads) ----
    v2f qa[8];
    {
      const float* qrow = qb + (size_t)lidx(qrow0 + l16, w) * DIM;
#pragma unroll
      for (int c = 0; c < 8; ++c) {
        v2f q2 = __builtin_nontemporal_load((const v2f*)(qrow + 4 * c + 2 * half));
        qa[c] = q2 * SCALE;
      }
    }

    // ---- S = Q * K^T  (16 col tiles, K=32 as 8 chained k4 WMMAs) ----
#pragma unroll 1
    for (int ct = 0; ct < 16; ++ct) {
      v8f acc = {};
      const float* kt = &Ks[(ct * 16 + l16) * KV_STRIDE + 2 * half];
#pragma unroll
      for (int c = 0; c < 8; ++c) {
        v2f bvv = *(const v2f*)(kt + 4 * c);
        acc = __builtin_amdgcn_wmma_f32_16x16x4_f32(
            false, qa[c], false, bvv, (short)0, acc, false, false);
      }
#pragma unroll
      for (int r = 0; r < 8; ++r)
        Ss[(r + 8 * half) * S_STRIDE + ct * 16 + l16] = acc[r];
    }

    // ---- bias add + softmax (lane owns half a row) ----
    const int r  = l16;
    const int cb = half * 128;
    float* srow  = Ss + r * S_STRIDE + cb;
    const int nh = (qrow0 + r) >> 2;
    const int nw = (qrow0 + r) & 3;
    float pmax = -3.0e38f;
#pragma unroll 4
    for (int c = 0; c < 128; ++c) {
      int m = cb + c;
      int idx = (nh - (m >> 2) + 63) * 7 + (nw - (m & 3) + 3);
      float s = srow[c] + ptab[idx * HEADS + hh];
      srow[c] = s;
      pmax = fmaxf(pmax, s);
    }
    pmax = fmaxf(pmax, __shfl_xor(pmax, 16, 32));
    float psum = 0.0f;
#pragma unroll 4
    for (int c = 0; c < 128; ++c) {
      float e = __expf(srow[c] - pmax);
      srow[c] = e;
      psum += e;
    }
    psum += __shfl_xor(psum, 16, 32);
    float inv = 1.0f / psum;      // lane l16 holds 1/rowsum for row l16

    // ---- O = P * V (K=256 as 64 chained k4 WMMAs), normalize on store ----
#pragma unroll 1
    for (int ct2 = 0; ct2 < 2; ++ct2) {
      v8f acc = {};
      const float* prow = &Ss[l16 * S_STRIDE + 2 * half];
      const float* vcol = &Vs[(2 * half) * KV_STRIDE + ct2 * 16 + l16];
#pragma unroll 4
      for (int c = 0; c < 64; ++c) {
        v2f av  = *(const v2f*)(prow + 4 * c);
        v2f bvv = { vcol[(4 * c) * KV_STRIDE], vcol[(4 * c + 1) * KV_STRIDE] };
        acc = __builtin_amdgcn_wmma_f32_16x16x4_f32(
            false, av, false, bvv, (short)0, acc, false, false);
      }
#pragma unroll
      for (int rr = 0; rr < 8; ++rr) {
        int orow = qrow0 + rr + 8 * half;
        float sc = __shfl(inv, rr + 8 * half, 32);
        __builtin_nontemporal_store(
            acc[rr] * sc,
            &out[(size_t)bb * 4096 * DIM + (size_t)lidx(orow, w) * DIM +
                 hh * HD + ct2 * 16 + l16]);
      }
    }
  }
}

// ---------------------------------------------------------------------------
extern "C" void kernel_launch(void* const* d_in, const int* in_sizes, int n_in,
                              void* d_out, int out_size, void* d_ws, size_t ws_size,
                              hipStream_t stream) {
  (void)in_sizes; (void)n_in; (void)out_size; (void)ws_size;
  const float* qkv = (const float*)d_in[0];
  const float* pw  = (const float*)d_in[1];
  const float* pb  = (const float*)d_in[2];
  const float* g1  = (const float*)d_in[3];
  const float* b1  = (const float*)d_in[4];
  const float* w1  = (const float*)d_in[5];
  const float* fb1 = (const float*)d_in[6];
  const float* g2  = (const float*)d_in[7];
  const float* b2  = (const float*)d_in[8];
  const float* w2  = (const float*)d_in[9];
  const float* fb2 = (const float*)d_in[10];
  const float* g3  = (const float*)d_in[11];
  const float* b3  = (const float*)d_in[12];
  const float* w3  = (const float*)d_in[13];
  const float* fb3 = (const float*)d_in[14];
  float* ptab = (float*)d_ws;
  float* out  = (float*)d_out;

  bias_mlp_kernel<<<dim3(7), dim3(128), 0, stream>>>(
      pw, pb, g1, b1, w1, fb1, g2, b2, w2, fb2, g3, b3, w3, fb3, ptab);

  const size_t smemBytes = (2 * KV_FLOATS + 4 * S_FLOATS) * sizeof(float); // 139,776 B
  hipFuncSetAttribute(reinterpret_cast<const void*>(attn_kernel),
                      hipFuncAttributeMaxDynamicSharedMemorySize, (int)smemBytes);
  attn_kernel<<<dim3(256, 6), dim3(128), smemBytes, stream>>>(qkv, ptab, out);
}